// GraphSAGE_GCN_17386027614491
// MI455X (gfx1250) — compile-verified
//
#include <hip/hip_runtime.h>
#include <math.h>

typedef __attribute__((ext_vector_type(2))) float v2f;
typedef __attribute__((ext_vector_type(8))) float v8f;

#define HID 128
#define TILE 16
#define BN_EPS 1e-5f

// ---------------------------------------------------------------------------
// Kernel 1: scatter-add of source features into per-destination accumulators.
// One wave32 per edge: each lane moves 4 floats (32*4 = 128 = HID).
// ---------------------------------------------------------------------------
__global__ void __launch_bounds__(256) sage_scatter_kernel(
    const float* __restrict__ x, const int* __restrict__ ei,
    float* __restrict__ agg, float* __restrict__ cnt, int nEdges)
{
    int gid  = blockIdx.x * blockDim.x + threadIdx.x;
    int e    = gid >> 5;
    int lane = gid & 31;
    if (e >= nEdges) return;
    int src = ei[e];
    int dst = ei[nEdges + e];
    const float4 v = ((const float4*)(x + (size_t)src * HID))[lane];
    float* a = agg + (size_t)dst * HID + lane * 4;
    unsafeAtomicAdd(a + 0, v.x);
    unsafeAtomicAdd(a + 1, v.y);
    unsafeAtomicAdd(a + 2, v.z);
    unsafeAtomicAdd(a + 3, v.w);
    if (lane == 0) unsafeAtomicAdd(cnt + dst, 1.0f);
}

// ---------------------------------------------------------------------------
// Kernel 2: h = (agg/cnt) @ W_l + b_l + x @ W_r  via V_WMMA_F32_16X16X4_F32.
// Block = 8 waves; wave w computes the 16x16 output tile (node tile = blockIdx,
// feature tile = w).  Also accumulates per-feature sum / sum-of-squares for BN.
// Requires nNodes % 16 == 0 (holds: 100000 = 6250*16) so EXEC is all ones.
// ---------------------------------------------------------------------------
__global__ void __launch_bounds__(256) sage_gemm_kernel(
    const float* __restrict__ agg, const float* __restrict__ xin,
    const float* __restrict__ Wl,  const float* __restrict__ bl,
    const float* __restrict__ Wr,  const float* __restrict__ cnt,
    float* __restrict__ h, float* __restrict__ gsum, float* __restrict__ gsq)
{
    __shared__ float ds_s[HID];
    __shared__ float ds_q[HID];

    const int lane = threadIdx.x & 31;
    const int wv   = threadIdx.x >> 5;      // feature tile 0..7
    const int mt   = blockIdx.x;            // node tile
    const int l15  = lane & 15;
    const int half = lane >> 4;             // 0: K pair {0,1}, 1: K pair {2,3}
    const int row  = mt * TILE + l15;       // A-matrix row owned by this lane
    const int ncol = wv * TILE + l15;       // B/C/D column owned by this lane

    if (threadIdx.x < HID) { ds_s[threadIdx.x] = 0.f; ds_q[threadIdx.x] = 0.f; }
    __syncthreads();

    const float inv  = 1.0f / fmaxf(cnt[row], 1.0f);   // mean-aggregation divide
    const float bias = bl[ncol];
    v8f c = { bias, bias, bias, bias, bias, bias, bias, bias };

    const float* arow = agg + (size_t)row * HID;
    const float* xrow = xin + (size_t)row * HID;

#pragma unroll
    for (int k0 = 0; k0 < HID; k0 += 4) {
        const int ka = k0 + half * 2;
        // A fragment (agg, scaled by 1/cnt) and matching W_l fragment
        v2f a;  a.x  = arow[ka] * inv;        a.y  = arow[ka + 1] * inv;
        v2f b;  b.x  = Wl[ka * HID + ncol];   b.y  = Wl[(ka + 1) * HID + ncol];
        c = __builtin_amdgcn_wmma_f32_16x16x4_f32(false, a, false, b,
                                                  (short)0, c, false, false);
        // A fragment (x) and matching W_r fragment
        v2f a2; a2.x = xrow[ka];              a2.y = xrow[ka + 1];
        v2f b2; b2.x = Wr[ka * HID + ncol];   b2.y = Wr[(ka + 1) * HID + ncol];
        c = __builtin_amdgcn_wmma_f32_16x16x4_f32(false, a2, false, b2,
                                                  (short)0, c, false, false);
    }

    // Store D tile + per-feature BN partial sums.
    // D layout: VGPR j -> row (mt*16 + half*8 + j), column ncol.
    float s = 0.f, q = 0.f;
    const int mbase = mt * TILE + half * 8;
#pragma unroll
    for (int j = 0; j < 8; ++j) {
        float v = c[j];
        h[(size_t)(mbase + j) * HID + ncol] = v;
        s += v;
        q += v * v;
    }
    atomicAdd(&ds_s[ncol], s);   // LDS ds_add_f32; 2 lanes/feature/wave
    atomicAdd(&ds_q[ncol], q);
    __syncthreads();
    if (threadIdx.x < HID) {
        unsafeAtomicAdd(gsum + threadIdx.x, ds_s[threadIdx.x]);
        unsafeAtomicAdd(gsq  + threadIdx.x, ds_q[threadIdx.x]);
    }
}

// ---------------------------------------------------------------------------
// Kernel 3: BatchNorm finalize + ELU, in place.
// ---------------------------------------------------------------------------
__global__ void __launch_bounds__(256) sage_bn_elu_kernel(
    float* __restrict__ h, const float* __restrict__ gsum,
    const float* __restrict__ gsq, const float* __restrict__ gamma,
    const float* __restrict__ beta, int total, float invN)
{
    int i = blockIdx.x * blockDim.x + threadIdx.x;
    if (i >= total) return;
    int f = i & (HID - 1);
    float mu  = gsum[f] * invN;
    float var = gsq[f] * invN - mu * mu;
    float v   = (h[i] - mu) * rsqrtf(var + BN_EPS) * gamma[f] + beta[f];
    h[i] = v > 0.f ? v : expm1f(v);
}

// ---------------------------------------------------------------------------
// Host-side launcher. Inputs (setup order): x, edge_index,
// then per layer: W_l, b_l, W_r, gamma, beta  (3 layers).
// ---------------------------------------------------------------------------
extern "C" void kernel_launch(void* const* d_in, const int* in_sizes, int n_in,
                              void* d_out, int out_size, void* d_ws, size_t ws_size,
                              hipStream_t stream)
{
    const float* x0 = (const float*)d_in[0];
    const int*   ei = (const int*)d_in[1];
    const int nEdges = in_sizes[1] / 2;
    const int nNodes = in_sizes[0] / HID;
    const size_t feat = (size_t)nNodes * HID;

    // Workspace carve-up: agg | xtmp | cnt | gsum | gsq
    float* agg  = (float*)d_ws;
    float* xtmp = agg + feat;
    float* cnt  = xtmp + feat;
    float* gsum = cnt + nNodes;
    float* gsq  = gsum + HID;

    float* out = (float*)d_out;

    // Ping-pong: L0: x0 -> d_out,  L1: d_out -> xtmp,  L2: xtmp -> d_out
    const float* lin[3]  = { x0, out, xtmp };
    float*       lout[3] = { out, xtmp, out };

    const int scatterBlocks = (nEdges + 7) / 8;        // 8 edges (waves) / block
    const int gemmBlocks    = nNodes / TILE;           // 6250, exact
    const int ewBlocks      = (int)((feat + 255) / 256);

    for (int layer = 0; layer < 3; ++layer) {
        const float* Wl    = (const float*)d_in[2 + 5 * layer + 0];
        const float* bl    = (const float*)d_in[2 + 5 * layer + 1];
        const float* Wr    = (const float*)d_in[2 + 5 * layer + 2];
        const float* gamma = (const float*)d_in[2 + 5 * layer + 3];
        const float* beta  = (const float*)d_in[2 + 5 * layer + 4];

        hipMemsetAsync(agg, 0, feat * sizeof(float), stream);
        hipMemsetAsync(cnt, 0, (size_t)nNodes * sizeof(float), stream);
        hipMemsetAsync(gsum, 0, 2 * HID * sizeof(float), stream); // gsum+gsq

        sage_scatter_kernel<<<scatterBlocks, 256, 0, stream>>>(
            lin[layer], ei, agg, cnt, nEdges);

        sage_gemm_kernel<<<gemmBlocks, 256, 0, stream>>>(
            agg, lin[layer], Wl, bl, Wr, cnt, lout[layer], gsum, gsq);

        sage_bn_elu_kernel<<<ewBlocks, 256, 0, stream>>>(
            lout[layer], gsum, gsq, gamma, beta, (int)feat,
            1.0f / (float)nNodes);
    }
}